// GenomicLogicBraidIntegrator_87385404605287
// MI455X (gfx1250) — compile-verified
//
#include <hip/hip_runtime.h>
#include <math.h>

typedef __attribute__((ext_vector_type(2))) float v2f;
typedef __attribute__((ext_vector_type(8))) float v8f;

#define NPTS 4096
#define EPSF 1e-9f

__device__ __forceinline__ void crossf(const float* a, const float* b, float* o) {
    o[0] = a[1]*b[2] - a[2]*b[1];
    o[1] = a[2]*b[0] - a[0]*b[2];
    o[2] = a[0]*b[1] - a[1]*b[0];
}

// Build per-segment packed feature vectors (8 floats each):
//  AN[i] = [ (m1 x dr1)(3), dr1(3), 0, 0 ]        BN[j] = [ dr2(3), -(dr2 x m2)(3), 0, 0 ]
//  AD[i] = [ m1(3), |m1|^2, 1, 0, 0, 0 ]          BD[j] = [ -2*m2(3), 1, |m2|^2, 0, 0, 0 ]
// so   AN.BN = (dr1 x dr2).(m1-m2)   and   AD.BD = |m1-m2|^2   (epsilon folded into WMMA C).
// Entry 4095 is zero-padded: AN row = 0 => num tile column/row contributes exactly 0.
__global__ void braid_prep_kernel(const float* __restrict__ bytes,
                                  const int*   __restrict__ idx,
                                  float* __restrict__ AN, float* __restrict__ AD,
                                  float* __restrict__ BN, float* __restrict__ BD) {
    int i = blockIdx.x * blockDim.x + threadIdx.x;
    if (i >= NPTS) return;

    float aN[8], aD[8], bN[8], bD[8];
#pragma unroll
    for (int k = 0; k < 8; ++k) { aN[k] = 0.f; aD[k] = 0.f; bN[k] = 0.f; bD[k] = 0.f; }

    if (i < NPTS - 1) {
        float r1a[3], r1b[3], r2a[3], r2b[3];
        // r1 = vector part of normalize([cos, .5 sin, .3 sin, .2 sin])
        {
            float th = bytes[i] * (float)(6.283185307179586 / 255.0);
            float c = cosf(th), s = sinf(th);
            float inv = rsqrtf(c*c + 0.38f*s*s + EPSF);
            r1a[0] = 0.5f*s*inv; r1a[1] = 0.3f*s*inv; r1a[2] = 0.2f*s*inv;
        }
        {
            float th = bytes[i+1] * (float)(6.283185307179586 / 255.0);
            float c = cosf(th), s = sinf(th);
            float inv = rsqrtf(c*c + 0.38f*s*s + EPSF);
            r1b[0] = 0.5f*s*inv; r1b[1] = 0.3f*s*inv; r1b[2] = 0.2f*s*inv;
        }
        // r2 = vector part of normalize(eye(4)[idx])
        {
            int k = idx[i];
            float inv = rsqrtf(1.0f + EPSF);
            r2a[0] = (k==1)?inv:0.f; r2a[1] = (k==2)?inv:0.f; r2a[2] = (k==3)?inv:0.f;
        }
        {
            int k = idx[i+1];
            float inv = rsqrtf(1.0f + EPSF);
            r2b[0] = (k==1)?inv:0.f; r2b[1] = (k==2)?inv:0.f; r2b[2] = (k==3)?inv:0.f;
        }

        float dr1[3], m1[3], dr2[3], m2[3];
#pragma unroll
        for (int k = 0; k < 3; ++k) {
            dr1[k] = r1b[k] - r1a[k];  m1[k] = 0.5f*(r1b[k] + r1a[k]);
            dr2[k] = r2b[k] - r2a[k];  m2[k] = 0.5f*(r2b[k] + r2a[k]);
        }
        float u[3], w[3];
        crossf(m1, dr1, u);    // (m1 x dr1)
        crossf(dr2, m2, w);    // (dr2 x m2)

        aN[0]=u[0]; aN[1]=u[1]; aN[2]=u[2]; aN[3]=dr1[0]; aN[4]=dr1[1]; aN[5]=dr1[2];
        bN[0]=dr2[0]; bN[1]=dr2[1]; bN[2]=dr2[2]; bN[3]=-w[0]; bN[4]=-w[1]; bN[5]=-w[2];

        float n1 = m1[0]*m1[0] + m1[1]*m1[1] + m1[2]*m1[2];
        float n2 = m2[0]*m2[0] + m2[1]*m2[1] + m2[2]*m2[2];
        aD[0]=m1[0]; aD[1]=m1[1]; aD[2]=m1[2]; aD[3]=n1; aD[4]=1.0f;
        bD[0]=-2.0f*m2[0]; bD[1]=-2.0f*m2[1]; bD[2]=-2.0f*m2[2]; bD[3]=1.0f; bD[4]=n2;
    }

#pragma unroll
    for (int k = 0; k < 8; ++k) {
        AN[i*8+k] = aN[k]; AD[i*8+k] = aD[k];
        BN[i*8+k] = bN[k]; BD[i*8+k] = bD[k];
    }
}

// One block per 16-row i-tile (256 blocks); 8 waves/block, each wave sweeps 32 j-tiles,
// reusing its A fragments across the sweep. 4x v_wmma_f32_16x16x4_f32 per tile.
__global__ void braid_tile_kernel(const float* __restrict__ AN, const float* __restrict__ AD,
                                  const float* __restrict__ BN, const float* __restrict__ BD,
                                  float* __restrict__ partials) {
    const int lane = threadIdx.x & 31;
    const int wave = threadIdx.x >> 5;
    const int M  = lane & 15;       // row (A) / col (B) within tile
    const int kh = lane >> 4;       // K half-select per ISA 16x4 layout
    const int i0 = blockIdx.x << 4;

    const float* arN = AN + (size_t)(i0 + M) * 8;
    const float* arD = AD + (size_t)(i0 + M) * 8;
    const v2f aNlo = *(const v2f*)(arN + 2*kh);
    const v2f aNhi = *(const v2f*)(arN + 4 + 2*kh);
    const v2f aDlo = *(const v2f*)(arD + 2*kh);
    const v2f aDhi = *(const v2f*)(arD + 4 + 2*kh);

    float acc = 0.0f;
    for (int jt = 0; jt < 32; ++jt) {
        const int j0 = ((wave << 5) + jt) << 4;
        const float* brN = BN + (size_t)(j0 + M) * 8;
        const float* brD = BD + (size_t)(j0 + M) * 8;
        const v2f bNlo = *(const v2f*)(brN + 2*kh);
        const v2f bNhi = *(const v2f*)(brN + 4 + 2*kh);
        const v2f bDlo = *(const v2f*)(brD + 2*kh);
        const v2f bDhi = *(const v2f*)(brD + 4 + 2*kh);

        v8f cn = {};
        cn = __builtin_amdgcn_wmma_f32_16x16x4_f32(false, aNlo, false, bNlo,
                                                   (short)0, cn, false, false);
        cn = __builtin_amdgcn_wmma_f32_16x16x4_f32(false, aNhi, false, bNhi,
                                                   (short)0, cn, false, false);
        v8f cd = {EPSF, EPSF, EPSF, EPSF, EPSF, EPSF, EPSF, EPSF};  // dist2 += EPS baked in
        cd = __builtin_amdgcn_wmma_f32_16x16x4_f32(false, aDlo, false, bDlo,
                                                   (short)0, cd, false, false);
        cd = __builtin_amdgcn_wmma_f32_16x16x4_f32(false, aDhi, false, bDhi,
                                                   (short)0, cd, false, false);

#pragma unroll
        for (int v = 0; v < 8; ++v) {
            // dist2 >= EPS = 1e-9 >> FLT_MIN, so the raw hardware v_rsq_f32 is safe:
            // skip libm's denormal-guard expansion entirely.
            float t = __builtin_amdgcn_rsqf(cd[v]);
            acc += cn[v] * (t * t * t);   // num * dist2^{-1.5}
        }
    }

    // deterministic in-block reduction
#pragma unroll
    for (int o = 16; o > 0; o >>= 1) acc += __shfl_xor(acc, o, 32);
    __shared__ float wsum[8];
    if (lane == 0) wsum[wave] = acc;
    __syncthreads();
    if (threadIdx.x == 0) {
        float s = 0.0f;
#pragma unroll
        for (int w = 0; w < 8; ++w) s += wsum[w];
        partials[blockIdx.x] = s;
    }
}

__global__ void braid_finalize_kernel(const float* __restrict__ partials,
                                      float* __restrict__ out) {
    const int t = threadIdx.x;
    float v = partials[t];
#pragma unroll
    for (int o = 16; o > 0; o >>= 1) v += __shfl_xor(v, o, 32);
    __shared__ float wsum[8];
    if ((t & 31) == 0) wsum[t >> 5] = v;
    __syncthreads();
    if (t == 0) {
        float s = 0.0f;
#pragma unroll
        for (int w = 0; w < 8; ++w) s += wsum[w];
        out[0] = s * 0.07957747154594767f;   // 1/(4*pi)
    }
}

extern "C" void kernel_launch(void* const* d_in, const int* in_sizes, int n_in,
                              void* d_out, int out_size, void* d_ws, size_t ws_size,
                              hipStream_t stream) {
    const float* bytes = (const float*)d_in[0];
    const int*   idx   = (const int*)d_in[1];

    float* ws = (float*)d_ws;
    float* AN = ws;                    // 4096*8 floats
    float* AD = ws + 1 * NPTS * 8;
    float* BN = ws + 2 * NPTS * 8;
    float* BD = ws + 3 * NPTS * 8;
    float* partials = ws + 4 * NPTS * 8;   // 256 floats

    braid_prep_kernel<<<NPTS / 256, 256, 0, stream>>>(bytes, idx, AN, AD, BN, BD);
    braid_tile_kernel<<<256, 256, 0, stream>>>(AN, AD, BN, BD, partials);
    braid_finalize_kernel<<<1, 256, 0, stream>>>(partials, (float*)d_out);
}